// CodeVulnerabilityGNN_36756330119704
// MI455X (gfx1250) — compile-verified
//
#include <hip/hip_runtime.h>
#include <hip/hip_bf16.h>

// Problem constants (match reference setup_inputs)
#define GNN_N      100000
#define GNN_E      1600000
#define GNN_FIN    10
#define GNN_H      128
#define GNN_L      3
#define GNN_G      256
#define GNN_C      25
#define GNN_NPAD   100096          // 782 * 128 : WMMA-tile padded row count
#define GNN_GEMM_BLOCKS 782

typedef float v2f __attribute__((ext_vector_type(2)));
typedef float v8f __attribute__((ext_vector_type(8)));

// ---------------- small utility kernels ----------------

__global__ __launch_bounds__(256) void gnn_zero_f4(float4* __restrict__ p, int n4) {
  int i = blockIdx.x * 256 + threadIdx.x;
  if (i < n4) p[i] = make_float4(0.f, 0.f, 0.f, 0.f);
}

__global__ __launch_bounds__(256) void gnn_deg_init(float* __restrict__ deg, int n) {
  int i = blockIdx.x * 256 + threadIdx.x;
  if (i < n) deg[i] = 1.0f;                       // self loop contributes 1
}

__global__ __launch_bounds__(256) void gnn_deg_acc(const int* __restrict__ rows,
                                                   float* __restrict__ deg, int e) {
  int i = blockIdx.x * 256 + threadIdx.x;
  if (i < e) atomicAdd(&deg[rows[i]], 1.0f);
}

__global__ __launch_bounds__(256) void gnn_rsqrt(float* __restrict__ d, int n) {
  int i = blockIdx.x * 256 + threadIdx.x;
  if (i < n) d[i] = rsqrtf(d[i]);
}

__global__ __launch_bounds__(256) void gnn_edge_norm(const int* __restrict__ rows,
                                                     const int* __restrict__ cols,
                                                     const float* __restrict__ dinv,
                                                     float* __restrict__ nrm, int e) {
  int i = blockIdx.x * 256 + threadIdx.x;
  if (i < e) nrm[i] = dinv[rows[i]] * dinv[cols[i]];
}

// ---------------- input projection: h = relu(x @ Wp + bp), K = 10 ----------------

__global__ __launch_bounds__(256) void gnn_proj(const float* __restrict__ x,
                                                const float* __restrict__ Wp,
                                                const float* __restrict__ bp,
                                                float* __restrict__ h) {
  int idx = blockIdx.x * 256 + threadIdx.x;
  if (idx >= GNN_N * GNN_H) return;
  int i = idx >> 7, j = idx & 127;
  float acc = bp[j];
#pragma unroll
  for (int k = 0; k < GNN_FIN; ++k)
    acc = fmaf(x[i * GNN_FIN + k], Wp[k * GNN_H + j], acc);
  h[(size_t)i * GNN_H + j] = fmaxf(acc, 0.f);
}

// ---------------- main GEMM: C[NPAD,128] = A[NPAD,128] @ W[128,128] ----------------
// fp32 WMMA 16x16x4.  W is staged in LDS in K-pair interleaved layout:
//   Wl[(k/2)*256 + n*2 + (k&1)] = W[k][n]
// so each B fragment (W[k][n], W[k+1][n]) is one contiguous 8-byte ds_load_b64
// landing in an aligned VGPR pair -> feeds v_wmma with no register shuffling.
// 8 waves/block, each wave: 16 rows x full 128 cols, 8 f32 accumulator tiles.
// The wave's entire A stripe (16x128 f32 = 64 VGPRs) is preloaded into registers
// so all 32 global_load_b64 issue back-to-back and drain under the compute.

__global__ __launch_bounds__(256) void gnn_gemm128(const float* __restrict__ Am,
                                                   const float* __restrict__ W,
                                                   float* __restrict__ Cm) {
  __shared__ float Wl[GNN_H * GNN_H];
  const int tid = threadIdx.x;
  {
    // idx enumerates (pair p, col n); global reads coalesced, LDS store is b64
#pragma unroll
    for (int i = 0; i < 32; ++i) {
      int idx = tid + i * 256;          // 0 .. 8191
      int pr  = idx >> 7;               // K pair 0..63
      int n   = idx & 127;
      v2f v;
      v.x = W[(pr * 2 + 0) * GNN_H + n];
      v.y = W[(pr * 2 + 1) * GNN_H + n];
      *(v2f*)(Wl + pr * 256 + n * 2) = v;
    }
  }

  const int wave = tid >> 5;
  const int lane = tid & 31;
  const int half = lane >> 4;     // 0: lanes 0-15, 1: lanes 16-31
  const int lm   = lane & 15;
  const size_t r0 = (size_t)blockIdx.x * 128 + (size_t)wave * 16;

  // A fragment preload: 16x4 f32 layout — lanes 0-15 hold K = {k0,k0+1},
  // lanes 16-31 hold K = {k0+2,k0+3}.  All 32 loads issue before any compute.
  const float* arow = Am + (r0 + lm) * GNN_H + half * 2;
  v2f afrag[32];
#pragma unroll
  for (int kk = 0; kk < 32; ++kk)
    afrag[kk] = *(const v2f*)(arow + kk * 4);

  v8f acc[8];
  const v8f vzero = {0.f, 0.f, 0.f, 0.f, 0.f, 0.f, 0.f, 0.f};
#pragma unroll
  for (int ct = 0; ct < 8; ++ct) acc[ct] = vzero;

  __syncthreads();   // W tile visible in LDS

#pragma unroll
  for (int kk = 0; kk < 32; ++kk) {
    // B fragment: k0 = 4*kk + 2*half -> pair index p = 2*kk + half, col n = 16*ct + lm
    const float* wb = Wl + (kk * 2 + half) * 256 + lm * 2;
#pragma unroll
    for (int ct = 0; ct < 8; ++ct) {
      v2f b = *(const v2f*)(wb + ct * 32);
      acc[ct] = __builtin_amdgcn_wmma_f32_16x16x4_f32(
          /*neg_a=*/false, afrag[kk], /*neg_b=*/false, b,
          /*c_mod=*/(short)0, acc[ct], /*reuse_a=*/false, /*reuse_b=*/false);
    }
  }

  // C/D layout: VGPR v, lanes 0-15 -> M=v, lanes 16-31 -> M=v+8
  float* crow = Cm + (r0 + half * 8) * GNN_H + lm;
#pragma unroll
  for (int ct = 0; ct < 8; ++ct)
#pragma unroll
    for (int v = 0; v < 8; ++v)
      crow[(size_t)v * GNN_H + ct * 16] = acc[ct][v];
}

// ---------------- edge scatter: agg[rows[e]] += hw[cols[e]] * norm[e] ----------------
// 32 lanes per edge, 4 features each; hw/agg are L2-resident (51 MB < 192 MB L2).

__global__ __launch_bounds__(256) void gnn_scatter(const int* __restrict__ rows,
                                                   const int* __restrict__ cols,
                                                   const float* __restrict__ nrm,
                                                   const float* __restrict__ hw,
                                                   float* __restrict__ agg) {
  int idx = blockIdx.x * 256 + threadIdx.x;
  int e = idx >> 5;
  if (e >= GNN_E) return;
  int f0 = (idx & 31) << 2;
  int r = rows[e], c = cols[e];
  float w = nrm[e];
  float4 v = *(const float4*)(hw + (size_t)c * GNN_H + f0);
  float* dst = agg + (size_t)r * GNN_H + f0;
  atomicAdd(dst + 0, v.x * w);
  atomicAdd(dst + 1, v.y * w);
  atomicAdd(dst + 2, v.z * w);
  atomicAdd(dst + 3, v.w * w);
}

// ---------------- self-loop + bias + BN(running stats) + ReLU (in place on agg) --------

__global__ __launch_bounds__(256) void gnn_bnrelu(float* __restrict__ agg,
                                                  const float* __restrict__ hw,
                                                  const float* __restrict__ dinv,
                                                  const float* __restrict__ bc,
                                                  const float* __restrict__ gamma,
                                                  const float* __restrict__ beta,
                                                  const float* __restrict__ mean,
                                                  const float* __restrict__ var) {
  int idx = blockIdx.x * 256 + threadIdx.x;
  if (idx >= GNN_N * GNN_H) return;
  int i = idx >> 7, j = idx & 127;
  float di = dinv[i];
  size_t o = (size_t)i * GNN_H + j;
  float t = agg[o] + hw[o] * di * di + bc[j];
  float s = gamma[j] * rsqrtf(var[j] + 1e-5f);
  float r = (t - mean[j]) * s + beta[j];
  agg[o] = fmaxf(r, 0.f);
}

// ---------------- pooling ----------------

__global__ __launch_bounds__(256) void gnn_pool_init(float* __restrict__ counts,
                                                     float* __restrict__ psum,
                                                     int* __restrict__ pmax) {
  int i = blockIdx.x * 256 + threadIdx.x;
  if (i < GNN_G * GNN_H) { psum[i] = 0.f; pmax[i] = 0; }
  if (i < GNN_G) counts[i] = 0.f;
}

__global__ __launch_bounds__(256) void gnn_count(const int* __restrict__ batch,
                                                 float* __restrict__ counts) {
  int i = blockIdx.x * 256 + threadIdx.x;
  if (i < GNN_N) atomicAdd(&counts[batch[i]], 1.0f);
}

__global__ __launch_bounds__(256) void gnn_pool(const float* __restrict__ h,
                                                const int* __restrict__ batch,
                                                float* __restrict__ psum,
                                                int* __restrict__ pmax) {
  int idx = blockIdx.x * 256 + threadIdx.x;
  int i = idx >> 5;
  if (i >= GNN_N) return;
  int f0 = (idx & 31) << 2;
  int b = batch[i];
  float4 v = *(const float4*)(h + (size_t)i * GNN_H + f0);
  float* ps = psum + (size_t)b * GNN_H + f0;
  int*   pm = pmax + (size_t)b * GNN_H + f0;
  atomicAdd(ps + 0, v.x); atomicAdd(ps + 1, v.y);
  atomicAdd(ps + 2, v.z); atomicAdd(ps + 3, v.w);
  // h >= 0 after ReLU -> non-negative float bit pattern is monotone under signed int compare
  atomicMax(pm + 0, __float_as_int(v.x)); atomicMax(pm + 1, __float_as_int(v.y));
  atomicMax(pm + 2, __float_as_int(v.z)); atomicMax(pm + 3, __float_as_int(v.w));
}

// ---------------- MLP head (tiny: scalar VALU is fine) ----------------

__global__ __launch_bounds__(256) void gnn_mlp1(const float* __restrict__ psum,
                                                const int* __restrict__ pmax,
                                                const float* __restrict__ counts,
                                                const float* __restrict__ W1,
                                                const float* __restrict__ b1,
                                                float* __restrict__ ghid) {
  int idx = blockIdx.x * 256 + threadIdx.x;
  if (idx >= GNN_G * GNN_H) return;
  int g = idx >> 7, j = idx & 127;
  float inv = 1.0f / fmaxf(counts[g], 1.0f);
  float acc = b1[j];
  for (int k = 0; k < GNN_H; ++k)
    acc = fmaf(psum[g * GNN_H + k] * inv, W1[k * GNN_H + j], acc);
  for (int k = 0; k < GNN_H; ++k)
    acc = fmaf(__int_as_float(pmax[g * GNN_H + k]), W1[(GNN_H + k) * GNN_H + j], acc);
  ghid[idx] = fmaxf(acc, 0.f);
}

__global__ __launch_bounds__(256) void gnn_mlp2(const float* __restrict__ ghid,
                                                const float* __restrict__ W2,
                                                const float* __restrict__ b2,
                                                float* __restrict__ out) {
  int idx = blockIdx.x * 256 + threadIdx.x;
  if (idx >= GNN_G * GNN_C) return;
  int g = idx / GNN_C, j = idx % GNN_C;
  float acc = b2[j];
  for (int k = 0; k < GNN_H; ++k)
    acc = fmaf(ghid[g * GNN_H + k], W2[k * GNN_C + j], acc);
  out[idx] = acc;
}

// ---------------- host launcher ----------------

extern "C" void kernel_launch(void* const* d_in, const int* in_sizes, int n_in,
                              void* d_out, int out_size, void* d_ws, size_t ws_size,
                              hipStream_t stream) {
  (void)in_sizes; (void)n_in; (void)out_size; (void)ws_size;

  const float* x     = (const float*)d_in[0];
  const int*   ei    = (const int*)d_in[1];
  const int*   rows  = ei;              // edge_index[0] : aggregation target
  const int*   cols  = ei + GNN_E;      // edge_index[1] : gather source
  const int*   batch = (const int*)d_in[2];
  const float* Wp    = (const float*)d_in[3];
  const float* bp    = (const float*)d_in[4];
  const float* Wc    = (const float*)d_in[5];
  const float* bc    = (const float*)d_in[6];
  const float* gamma = (const float*)d_in[7];
  const float* beta  = (const float*)d_in[8];
  const float* mean  = (const float*)d_in[9];
  const float* var   = (const float*)d_in[10];
  const float* W1    = (const float*)d_in[11];
  const float* b1    = (const float*)d_in[12];
  const float* W2    = (const float*)d_in[13];
  const float* b2    = (const float*)d_in[14];
  float*       out   = (float*)d_out;

  // Workspace layout (~110 MB): two ping-pong node-feature buffers + edge data
  char* p = (char*)d_ws;
  auto alignup = [](size_t v) { return (v + 255) & ~(size_t)255; };
  size_t off = 0;
  float* bufA  = (float*)(p + off); off = alignup(off + (size_t)GNN_NPAD * GNN_H * 4); // h / agg
  float* bufB  = (float*)(p + off); off = alignup(off + (size_t)GNN_NPAD * GNN_H * 4); // hw
  float* dinv  = (float*)(p + off); off = alignup(off + (size_t)GNN_N * 4);
  float* enorm = (float*)(p + off); off = alignup(off + (size_t)GNN_E * 4);
  float* cnts  = (float*)(p + off); off = alignup(off + (size_t)GNN_G * 4);
  float* psum  = (float*)(p + off); off = alignup(off + (size_t)GNN_G * GNN_H * 4);
  int*   pmax  = (int*)  (p + off); off = alignup(off + (size_t)GNN_G * GNN_H * 4);
  float* ghid  = (float*)(p + off); off = alignup(off + (size_t)GNN_G * GNN_H * 4);

  const int T = 256;
  const int gN   = (GNN_N + T - 1) / T;
  const int gE   = (GNN_E + T - 1) / T;
  const int gNH  = (GNN_N * GNN_H + T - 1) / T;
  const int gPadZ = (GNN_NPAD * GNN_H / 4 + T - 1) / T;
  const int gE32 = (int)(((size_t)GNN_E * 32 + T - 1) / T);
  const int gN32 = (int)(((size_t)GNN_N * 32 + T - 1) / T);
  const int gGH  = (GNN_G * GNN_H + T - 1) / T;
  const int gGC  = (GNN_G * GNN_C + T - 1) / T;

  // Degree + symmetric normalization
  gnn_deg_init<<<gN, T, 0, stream>>>(dinv, GNN_N);
  gnn_deg_acc<<<gE, T, 0, stream>>>(rows, dinv, GNN_E);
  gnn_rsqrt<<<gN, T, 0, stream>>>(dinv, GNN_N);
  gnn_edge_norm<<<gE, T, 0, stream>>>(rows, cols, dinv, enorm, GNN_E);

  // Input projection (pad rows get zeroed by the per-layer zero below)
  gnn_proj<<<gNH, T, 0, stream>>>(x, Wp, bp, bufA);

  for (int l = 0; l < GNN_L; ++l) {
    gnn_gemm128<<<GNN_GEMM_BLOCKS, T, 0, stream>>>(bufA, Wc + (size_t)l * GNN_H * GNN_H, bufB);
    gnn_zero_f4<<<gPadZ, T, 0, stream>>>((float4*)bufA, GNN_NPAD * GNN_H / 4);
    gnn_scatter<<<gE32, T, 0, stream>>>(rows, cols, enorm, bufB, bufA);
    gnn_bnrelu<<<gNH, T, 0, stream>>>(bufA, bufB, dinv,
                                      bc + l * GNN_H, gamma + l * GNN_H, beta + l * GNN_H,
                                      mean + l * GNN_H, var + l * GNN_H);
  }

  // Global mean/max pooling + MLP head
  gnn_pool_init<<<gGH, T, 0, stream>>>(cnts, psum, pmax);
  gnn_count<<<gN, T, 0, stream>>>(batch, cnts);
  gnn_pool<<<gN32, T, 0, stream>>>(bufA, batch, psum, pmax);
  gnn_mlp1<<<gGH, T, 0, stream>>>(psum, pmax, cnts, W1, b1, ghid);
  gnn_mlp2<<<gGC, T, 0, stream>>>(ghid, W2, b2, out);
}